// CBAMAttentionModule_90606630076539
// MI455X (gfx1250) — compile-verified
//
#include <hip/hip_runtime.h>
#include <hip/hip_bf16.h>

typedef __attribute__((ext_vector_type(2))) float v2f;
typedef __attribute__((ext_vector_type(4))) float v4f;
typedef __attribute__((ext_vector_type(8))) float v8f;

#define B_ 16
#define C_ 1024
#define N_ 16384
#define R_ 128           // C / RATIO
#define BN_EPS 1e-5f

// ---------------------------------------------------------------------------
// K1: per-(b,c) mean and max over N.  One block per row of 16384 floats.
// x stream is single-use -> non-temporal loads (don't pollute L2).
// ---------------------------------------------------------------------------
__global__ void pool_n_kernel(const float* __restrict__ x,
                              float* __restrict__ m_out,
                              float* __restrict__ mx_out) {
    const int row = blockIdx.x;                 // b*C + c
    const int tid = threadIdx.x;                // 0..255
    const v4f* xr = (const v4f*)(x + (size_t)row * N_);
    float s = 0.0f;
    float mv = -__builtin_inff();
    #pragma unroll 4
    for (int i = tid; i < N_ / 4; i += 256) {
        v4f v = __builtin_nontemporal_load(xr + i);
        s += (v.x + v.y) + (v.z + v.w);
        mv = fmaxf(mv, fmaxf(fmaxf(v.x, v.y), fmaxf(v.z, v.w)));
    }
    __shared__ float ssum[256];
    __shared__ float smax[256];
    ssum[tid] = s; smax[tid] = mv;
    __syncthreads();
    for (int off = 128; off > 0; off >>= 1) {
        if (tid < off) {
            ssum[tid] += ssum[tid + off];
            smax[tid] = fmaxf(smax[tid], smax[tid + off]);
        }
        __syncthreads();
    }
    if (tid == 0) {
        m_out[row]  = ssum[0] * (1.0f / (float)N_);
        mx_out[row] = smax[0];
    }
}

// ---------------------------------------------------------------------------
// WMMA fp32 16x16x4 layouts (ISA 7.12.2):
//   A 16x4 : lane L (L<16): M=L, VGPR0=K0,VGPR1=K1 ; lanes 16-31: K2,K3
//   B 4x16 : lane L: N=L&15, lane-group selects K pair (K0/K1 vs K2/K3)
//   C/D    : VGPR i, lane L: M = i + 8*(L>>4), N = L&15
// ---------------------------------------------------------------------------

// K2: h[src][m][r] = relu( sum_c v_src[m][c] * w1[r][c] ), M=16, N(r)=128, K=1024
// grid = (8 r-tiles, 2 sources), 32 threads (one wave, EXEC all ones).
__global__ void mlp1_wmma_kernel(const float* __restrict__ m_in,
                                 const float* __restrict__ mx_in,
                                 const float* __restrict__ w1,
                                 float* __restrict__ h_out) {
    const int lane  = threadIdx.x;                  // 0..31
    const int ntile = blockIdx.x;                   // 0..7
    const float* v  = (blockIdx.y == 0) ? m_in : mx_in;
    float* hout     = h_out + (size_t)blockIdx.y * (B_ * R_);

    const int mrow  = lane & 15;
    const int khalf = (lane >> 4) << 1;             // 0 or 2
    const int ncol  = ntile * 16 + (lane & 15);     // r index for B frag

    v8f acc = {};
    for (int kb = 0; kb < C_; kb += 4) {
        v2f a, b;
        a.x = v[mrow * C_ + kb + khalf];
        a.y = v[mrow * C_ + kb + khalf + 1];
        b.x = w1[ncol * C_ + kb + khalf];           // B[k][n] = w1[n*C + k]
        b.y = w1[ncol * C_ + kb + khalf + 1];
        acc = __builtin_amdgcn_wmma_f32_16x16x4_f32(
            false, a, false, b, (short)0, acc, false, false);
    }
    const int rowbase = (lane >> 4) * 8;
    const int col     = ntile * 16 + (lane & 15);
    #pragma unroll
    for (int i = 0; i < 8; ++i) {
        int row = i + rowbase;                      // batch index m
        hout[row * R_ + col] = fmaxf(acc[i], 0.0f); // relu
    }
}

// K3: gate[m][c] = sigmoid( sum_r w2[c][r]*(h_m[m][r] + h_mx[m][r]) )
// M=16, N(c)=1024, K=128, accumulated over both sources. grid = 64 tiles.
__global__ void mlp2_wmma_kernel(const float* __restrict__ h_in,
                                 const float* __restrict__ w2,
                                 float* __restrict__ gate) {
    const int lane  = threadIdx.x;
    const int ctile = blockIdx.x;                   // 0..63
    const int mrow  = lane & 15;
    const int khalf = (lane >> 4) << 1;
    const int ncol  = ctile * 16 + (lane & 15);     // c index

    v8f acc = {};
    for (int src = 0; src < 2; ++src) {
        const float* hs = h_in + (size_t)src * (B_ * R_);
        for (int kb = 0; kb < R_; kb += 4) {
            v2f a, b;
            a.x = hs[mrow * R_ + kb + khalf];
            a.y = hs[mrow * R_ + kb + khalf + 1];
            b.x = w2[ncol * R_ + kb + khalf];       // B[k][n] = w2[n*R + k]
            b.y = w2[ncol * R_ + kb + khalf + 1];
            acc = __builtin_amdgcn_wmma_f32_16x16x4_f32(
                false, a, false, b, (short)0, acc, false, false);
        }
    }
    const int rowbase = (lane >> 4) * 8;
    #pragma unroll
    for (int i = 0; i < 8; ++i) {
        int row = i + rowbase;                      // batch index
        float y = acc[i];
        gate[row * C_ + ctile * 16 + (lane & 15)] = 1.0f / (1.0f + __expf(-y));
    }
}

// ---------------------------------------------------------------------------
// K4: per-(b,n) mean/max over C of gate[b][c]*x[b][c][n]. Streaming pass.
// grid = (N/1024, B), 256 threads, each thread owns one v4f (4 n values).
// x loads non-temporal; gate is L2/SGPR-resident.
// ---------------------------------------------------------------------------
__global__ void spatial_pool_kernel(const float* __restrict__ x,
                                    const float* __restrict__ gate,
                                    float* __restrict__ cm,
                                    float* __restrict__ cx) {
    const int b  = blockIdx.y;
    const int n4 = blockIdx.x * 256 + threadIdx.x;      // v4f index in N/4
    const v4f* x4 = (const v4f*)x;
    const size_t base = (size_t)b * C_ * (N_ / 4);
    const float* g = gate + b * C_;

    v4f s  = {0.f, 0.f, 0.f, 0.f};
    float ninf = -__builtin_inff();
    v4f mv = {ninf, ninf, ninf, ninf};
    #pragma unroll 4
    for (int c = 0; c < C_; ++c) {
        float gc = g[c];                                // uniform -> scalar load
        v4f v = __builtin_nontemporal_load(x4 + base + (size_t)c * (N_ / 4) + n4);
        float a0 = gc * v.x, a1 = gc * v.y, a2 = gc * v.z, a3 = gc * v.w;
        s.x += a0; s.y += a1; s.z += a2; s.w += a3;
        mv.x = fmaxf(mv.x, a0); mv.y = fmaxf(mv.y, a1);
        mv.z = fmaxf(mv.z, a2); mv.w = fmaxf(mv.w, a3);
    }
    const int out = b * (N_ / 4) + n4;
    const float sc = 1.0f / (float)C_;
    v4f sm = {s.x * sc, s.y * sc, s.z * sc, s.w * sc};
    ((v4f*)cm)[out] = sm;    // small + re-read soon -> keep RT (L2 resident)
    ((v4f*)cx)[out] = mv;
}

// ---------------------------------------------------------------------------
// K5: s = relu(BN(sw0*cx + sw1*cm)); att = softmax_N(s). One block per batch.
// cm/cx are 2 MB total -> L2 resident across the three passes.
// ---------------------------------------------------------------------------
__global__ void softmax_n_kernel(const float* __restrict__ cm,
                                 const float* __restrict__ cx,
                                 const float* __restrict__ sw,
                                 const float* __restrict__ gamma,
                                 const float* __restrict__ beta,
                                 const float* __restrict__ rmean,
                                 const float* __restrict__ rvar,
                                 float* __restrict__ att) {
    const int b   = blockIdx.x;
    const int tid = threadIdx.x;
    const float sw0 = sw[0], sw1 = sw[1];
    const float inv = rsqrtf(rvar[0] + BN_EPS);
    const float ga = gamma[0], be = beta[0], mu = rmean[0];
    const float* cmb = cm + b * N_;
    const float* cxb = cx + b * N_;

    __shared__ float red[256];

    float lmax = -__builtin_inff();
    for (int n = tid; n < N_; n += 256) {
        float raw = sw0 * cxb[n] + sw1 * cmb[n];
        float t = fmaxf((raw - mu) * inv * ga + be, 0.0f);
        lmax = fmaxf(lmax, t);
    }
    red[tid] = lmax; __syncthreads();
    for (int off = 128; off > 0; off >>= 1) {
        if (tid < off) red[tid] = fmaxf(red[tid], red[tid + off]);
        __syncthreads();
    }
    const float gmax = red[0];
    __syncthreads();

    float lsum = 0.0f;
    for (int n = tid; n < N_; n += 256) {
        float raw = sw0 * cxb[n] + sw1 * cmb[n];
        float t = fmaxf((raw - mu) * inv * ga + be, 0.0f);
        lsum += __expf(t - gmax);
    }
    red[tid] = lsum; __syncthreads();
    for (int off = 128; off > 0; off >>= 1) {
        if (tid < off) red[tid] += red[tid + off];
        __syncthreads();
    }
    const float rsum = 1.0f / red[0];

    for (int n = tid; n < N_; n += 256) {
        float raw = sw0 * cxb[n] + sw1 * cmb[n];
        float t = fmaxf((raw - mu) * inv * ga + be, 0.0f);
        att[b * N_ + n] = __expf(t - gmax) * rsum;
    }
}

// ---------------------------------------------------------------------------
// K6: out = att[b][n] * gate[b][c] * x[b][c][n].  v4f streaming.
// Each block covers 256 v4f inside one (b,c) row -> b,c uniform (scalar loads).
// x load + out store non-temporal; att (1 MB, reused 1024x) stays RT in L2.
// ---------------------------------------------------------------------------
__global__ void apply_kernel(const float* __restrict__ x,
                             const float* __restrict__ gate,
                             const float* __restrict__ att,
                             float* __restrict__ out) {
    const int row    = blockIdx.x >> 4;             // b*C + c   (16 blocks/row)
    const int within = blockIdx.x & 15;
    const int b = row >> 10;
    const float g = gate[row];                      // uniform -> scalar load
    const int n4 = within * 256 + threadIdx.x;      // v4f index in N/4
    const size_t idx = (size_t)row * (N_ / 4) + n4;

    v4f v = __builtin_nontemporal_load((const v4f*)x + idx);
    v4f a = ((const v4f*)att)[(size_t)b * (N_ / 4) + n4];
    v4f o;
    o.x = g * a.x * v.x;
    o.y = g * a.y * v.y;
    o.z = g * a.z * v.z;
    o.w = g * a.w * v.w;
    __builtin_nontemporal_store(o, (v4f*)out + idx);
}

// ---------------------------------------------------------------------------
extern "C" void kernel_launch(void* const* d_in, const int* in_sizes, int n_in,
                              void* d_out, int out_size, void* d_ws, size_t ws_size,
                              hipStream_t stream) {
    const float* x     = (const float*)d_in[0];
    const float* w1    = (const float*)d_in[1];
    const float* w2    = (const float*)d_in[2];
    const float* sw    = (const float*)d_in[3];
    const float* gamma = (const float*)d_in[4];
    const float* beta  = (const float*)d_in[5];
    const float* rmean = (const float*)d_in[6];
    const float* rvar  = (const float*)d_in[7];
    float* out = (float*)d_out;

    // workspace carve-out (floats), ~3.4 MB total
    float* ws   = (float*)d_ws;
    float* m    = ws;                         // B*C       = 16384
    float* mx   = m    + B_ * C_;             // 16384
    float* h    = mx   + B_ * C_;             // 2*B*R     = 4096
    float* gate = h    + 2 * B_ * R_;         // B*C       = 16384
    float* cm   = gate + B_ * C_;             // B*N       = 262144
    float* cx   = cm   + B_ * N_;             // 262144
    float* att  = cx   + B_ * N_;             // 262144

    // 1) channel pools over N
    pool_n_kernel<<<B_ * C_, 256, 0, stream>>>(x, m, mx);

    // 2) MLP layer 1 (WMMA f32 16x16x4), both pooled vectors
    mlp1_wmma_kernel<<<dim3(R_ / 16, 2), 32, 0, stream>>>(m, mx, w1, h);

    // 3) MLP layer 2 (WMMA) + sigmoid -> gate
    mlp2_wmma_kernel<<<C_ / 16, 32, 0, stream>>>(h, w2, gate);

    // 4) spatial pools over C of gate*x
    spatial_pool_kernel<<<dim3(N_ / 1024, B_), 256, 0, stream>>>(x, gate, cm, cx);

    // 5) BN + relu + softmax over N
    softmax_n_kernel<<<B_, 256, 0, stream>>>(cm, cx, sw, gamma, beta, rmean, rvar, att);

    // 6) out = att * gate * x
    apply_kernel<<<(B_ * C_ * N_ / 4) / 256, 256, 0, stream>>>(x, gate, att, out);
}